// CNN_noninvariant_plaq_28226525069676
// MI455X (gfx1250) — compile-verified
//
#include <hip/hip_runtime.h>
#include <cstdint>
#include <cstddef>

// Problem constants (reference: O,I,K,N = 32,32,13,4096)
#define O_DIM 32
#define I_DIM 32
#define K_DIM 13
#define N_DIM 4096
#define IK    (I_DIM * K_DIM)   // 416 floats per (n,o) mask row / per W row

typedef float v4f __attribute__((ext_vector_type(4)));

// y[o,n] = sigmoid-ish( b[o] + sum_{i,k} W[o,i,k] * mask[n,o,i,k] * x[i, shifts[n,k]] )
// Bandwidth-bound on mask (218 MB, single use) -> stream NT; W cached in LDS via
// CDNA5 async global->LDS loads; g gather cached in LDS.
__global__ __launch_bounds__(256) void plaq_stream_kernel(
    const float* __restrict__ x,       // (I, N)
    const float* __restrict__ W,       // (O, I, K) = (O, 416)
    const float* __restrict__ b,       // (O,)
    const float* __restrict__ mask,    // (N, O, I, K)
    const int*   __restrict__ shifts,  // (N, K)
    float* __restrict__ out)           // (O, N)
{
    __shared__ __align__(16) float w_lds[O_DIM * IK]; // 53248 B
    __shared__ __align__(16) float g_lds[IK];         // 1664 B

    const int n    = blockIdx.x;
    const int tid  = threadIdx.x;
    const int lane = tid & 31;
    const int wave = tid >> 5;

    // ---- Stage W into LDS with async global->LDS b128 copies ----------------
    // 53248 B total = 104 chunks of 512 B (32 lanes x 16 B). 8 waves x 13 chunks.
    {
        const uint32_t lds_base = (uint32_t)(uintptr_t)(&w_lds[0]);
        #pragma unroll
        for (int c = 0; c < 13; ++c) {
            uint32_t off  = ((uint32_t)(wave * 13 + c) * 512u) + (uint32_t)lane * 16u;
            uint32_t ldsa = lds_base + off;
            // GVS mode: mem = SADDR(64b) + VADDR(32b offset); LDS dest addr in VDST VGPR.
            asm volatile("global_load_async_to_lds_b128 %0, %1, %2"
                         :
                         : "v"(ldsa), "v"(off), "s"(W)
                         : "memory");
        }
    }

    // ---- Stage gathered g[e] = x[i, shifts[n,k]] (e = i*13 + k) -------------
    for (int e = tid; e < IK; e += 256) {
        int i = e / K_DIM;
        int k = e - i * K_DIM;
        int s = shifts[n * K_DIM + k];
        g_lds[e] = x[i * N_DIM + s];
    }

    // Wait for this wave's async LDS writes, then block barrier so every wave
    // sees the full W slab + g gather.
    asm volatile("s_wait_asynccnt 0" ::: "memory");
    __syncthreads();

    const float* mrow_base = mask + (size_t)n * O_DIM * IK;

    const float e_const = 2.718281828459045f;
    const float scale   = (2.0f + 2.0f * e_const) / (e_const - 1.0f);

    // Each wave handles 4 o-rows (8 waves x 4 = 32 = O_DIM).
    #pragma unroll
    for (int oo = 0; oo < 4; ++oo) {
        const int o = wave * 4 + oo;
        const float* __restrict__ mrow = mrow_base + (size_t)o * IK;
        const float* __restrict__ wrow = &w_lds[o * IK];

        float acc = 0.0f;

        // 3 x (32 lanes x float4) = 384 elements, fully coalesced b128 NT loads.
        #pragma unroll
        for (int t = 0; t < 3; ++t) {
            const int e4 = (t * 32 + lane) * 4;
            v4f m = __builtin_nontemporal_load((const v4f*)(mrow + e4));
            v4f wv = *(const v4f*)(wrow + e4);
            v4f gv = *(const v4f*)(&g_lds[e4]);
            acc += m.x * wv.x * gv.x;
            acc += m.y * wv.y * gv.y;
            acc += m.z * wv.z * gv.z;
            acc += m.w * wv.w * gv.w;
        }
        // Tail: elements 384..415 (one b32 per lane).
        {
            const int e = 384 + lane;
            float m = __builtin_nontemporal_load(mrow + e);
            acc += m * wrow[e] * g_lds[e];
        }

        // Wave32 butterfly reduction.
        #pragma unroll
        for (int s = 16; s > 0; s >>= 1)
            acc += __shfl_xor(acc, s, 32);

        if (lane == 0) {
            float y = acc + b[o];
            float r = (1.0f / (1.0f + __expf(-y)) - 0.5f) * scale;
            out[(size_t)o * N_DIM + n] = r;
        }
    }
}

extern "C" void kernel_launch(void* const* d_in, const int* in_sizes, int n_in,
                              void* d_out, int out_size, void* d_ws, size_t ws_size,
                              hipStream_t stream) {
    (void)in_sizes; (void)n_in; (void)out_size; (void)d_ws; (void)ws_size;

    const float* x      = (const float*)d_in[0];  // (32, 4096)
    const float* Wconv  = (const float*)d_in[1];  // (32, 32, 13)
    const float* bconv  = (const float*)d_in[2];  // (32,)
    const float* mask   = (const float*)d_in[3];  // (4096, 32, 32, 13)
    const int*   shifts = (const int*)d_in[4];    // (4096, 13)
    float*       out    = (float*)d_out;          // (32, 4096)

    plaq_stream_kernel<<<N_DIM, 256, 0, stream>>>(x, Wconv, bconv, mask, shifts, out);
}